// OSTL_67181878444190
// MI455X (gfx1250) — compile-verified
//
#include <hip/hip_runtime.h>

typedef __attribute__((ext_vector_type(2))) float v2f;
typedef __attribute__((ext_vector_type(4))) float v4f;
typedef __attribute__((ext_vector_type(8))) float v8f;

#define SIG_TAU 0.8807970779778823f   // sigmoid(2.0)
#define IN_SZ   8192
#define OUT_SZ  8192

// ---------------------------------------------------------------------------
// Kernel A: I = x @ W via V_WMMA_F32_16X16X4_F32, then LIF cell update.
// Grid: 512 blocks (16 output columns each) x 256 threads (8 waves).
// Each wave owns a K-slice of 1024 (256 WMMAs). A is filled with x[k] in
// EVERY row (A[m,k] = x[k]) so all 16 rows of D carry the same partial I:
// redundant FLOPs are free (memory-bound on W), and the fragment build is a
// single uniform ds_load_b64 per lane -- no EXEC divergence in the hot loop.
// Per the ISA 32-bit A 16x4 layout, lane L holds A[L%16, K] with
// K = {0,1} (VGPR0,VGPR1) for lanes 0-15 and K = {2,3} for lanes 16-31.
// ---------------------------------------------------------------------------
__global__ __launch_bounds__(256) void ostl_matvec_lif(
    const float* __restrict__ x,
    const float* __restrict__ u,
    const float* __restrict__ W,
    float* __restrict__ u_out,
    float* __restrict__ s_out)
{
    __shared__ __align__(16) float xs[IN_SZ];
    __shared__ float partial[8][16];

    const int tid = threadIdx.x;

    // Stage x into LDS (2048 v4f / 256 threads = 8 each)
    const v4f* x4  = (const v4f*)x;
    v4f*       xs4 = (v4f*)xs;
    #pragma unroll
    for (int i = 0; i < 8; ++i) xs4[tid + i * 256] = x4[tid + i * 256];
    __syncthreads();

    const int wave  = tid >> 5;        // 0..7
    const int lane  = tid & 31;
    const int laneN = lane & 15;
    const int n0    = blockIdx.x * 16; // output column tile
    const int kBase = wave * 1024;     // K-slice for this wave

    // A fragment source: lanes 0-15 read xs[k], xs[k+1]; lanes 16-31 read
    // xs[k+2], xs[k+3]  (8-byte aligned -> ds_load_b64)
    const float* xp = xs + ((lane >> 4) << 1);

    // B fragment: lanes 0-15 carry K rows {0,1}, lanes 16-31 carry K rows {2,3}
    const int rowOff = (lane < 16) ? 0 : 2;
    const float* wp = W + (size_t)(kBase + rowOff) * OUT_SZ + n0 + laneN;

    v8f c = {0.f, 0.f, 0.f, 0.f, 0.f, 0.f, 0.f, 0.f};

    #pragma unroll 4
    for (int k = kBase; k < kBase + 1024; k += 4) {
        v2f A = *(const v2f*)(xp + k);          // uniform ds_load_b64

        v2f B; B.x = wp[0]; B.y = wp[OUT_SZ];   // W[k+rowOff, n], W[k+rowOff+1, n]
        wp += 4 * OUT_SZ;

        c = __builtin_amdgcn_wmma_f32_16x16x4_f32(
                /*neg_a=*/false, A, /*neg_b=*/false, B,
                /*c_mod=*/(short)0, c, /*reuse_a=*/false, /*reuse_b=*/false);
    }

    // Every row of D is identical; take row M=0: c[0], lanes 0-15 (N = lane)
    if (lane < 16) partial[wave][lane] = c[0];
    __syncthreads();

    if (wave == 0 && lane < 16) {
        float I = 0.f;
        #pragma unroll
        for (int w2 = 0; w2 < 8; ++w2) I += partial[w2][lane];
        const int n = n0 + lane;
        const float un = SIG_TAU * u[n] + I;
        const float sp = (un - 1.0f > 0.0f) ? 1.0f : 0.0f;  // V_TH = 1
        u_out[n] = un - sp;                                  // (V_TH - V_RESET) = 1
        s_out[n] = sp;
    }
}

// ---------------------------------------------------------------------------
// Kernel B: J' = sig_tau * J + x[row], pure streaming (512 MB of traffic).
// Non-temporal hints keep the one-shot J stream from polluting L2 (reserve
// L2 for W reuse). Grid: (8192 rows, 2 chunks) x 256 threads, 4 v4f each.
// ---------------------------------------------------------------------------
__global__ __launch_bounds__(256) void ostl_trace(
    const float* __restrict__ x,
    const float* __restrict__ J,
    float* __restrict__ Jout)
{
    const int row   = blockIdx.x;
    const int chunk = blockIdx.y;
    const float xr  = x[row];  // uniform -> scalar load

    const size_t base = (size_t)row * OUT_SZ + (size_t)chunk * 4096
                      + (size_t)threadIdx.x * 4;
    const v4f* Jp = (const v4f*)(J + base);
    v4f*       Op = (v4f*)(Jout + base);

    #pragma unroll
    for (int i = 0; i < 4; ++i) {
        v4f j = __builtin_nontemporal_load(&Jp[i * 256]);
        v4f o;
        o.x = SIG_TAU * j.x + xr;
        o.y = SIG_TAU * j.y + xr;
        o.z = SIG_TAU * j.z + xr;
        o.w = SIG_TAU * j.w + xr;
        __builtin_nontemporal_store(o, &Op[i * 256]);
    }
}

// ---------------------------------------------------------------------------
extern "C" void kernel_launch(void* const* d_in, const int* in_sizes, int n_in,
                              void* d_out, int out_size, void* d_ws, size_t ws_size,
                              hipStream_t stream) {
    (void)in_sizes; (void)n_in; (void)out_size; (void)d_ws; (void)ws_size;

    const float* x = (const float*)d_in[0];   // [8192]
    const float* u = (const float*)d_in[1];   // [8192]
    const float* J = (const float*)d_in[2];   // [8192, 8192]
    const float* W = (const float*)d_in[3];   // [8192, 8192]

    float* out    = (float*)d_out;
    float* u_out  = out;                                   // [8192]
    float* J_out  = out + OUT_SZ;                          // [8192*8192]
    float* s_out  = out + OUT_SZ + (size_t)IN_SZ * OUT_SZ; // [8192]

    ostl_matvec_lif<<<OUT_SZ / 16, 256, 0, stream>>>(x, u, W, u_out, s_out);
    ostl_trace<<<dim3(IN_SZ, 2), 256, 0, stream>>>(x, J, J_out);
}